// CombinedLoss_24575802868185
// MI455X (gfx1250) — compile-verified
//
#include <hip/hip_runtime.h>
#include <math.h>

typedef __attribute__((ext_vector_type(2))) float v2f;
typedef __attribute__((ext_vector_type(4))) float v4f;
typedef __attribute__((ext_vector_type(8))) float v8f;

// ---------------------------------------------------------------------------
// Shapes (reference setup_inputs): B=8, M=2048, S=512, N=256
//   pc1_0 : [8,2048,3] -> 16384 pts   pc1_1 : [8,512,3] -> 4096 pts
//   pc1_3 : [8,256,1]                 pc2   : [8,2048,3] -> 16384 pts
//   pc3   : [8,256,3]  -> 2048 pts (batched 256 x 2048 NN problem)
//
// d2_ij = ||a_i||^2 + ||b_j||^2 - 2 a_i.b_j  computed 16x16 per
// V_WMMA_F32_16X16X4_F32 with A_i=(-2ax,-2ay,-2az,1), B_j=(bx,by,bz,||b||^2),
// C=0;  ||a_i||^2 folded in after the column-min (min commutes with +const
// per row, and max(.,0) commutes with min).
//
// Workspace (floats):
//   [0..7]                accumulators
//   PACK_A0: 65536 floats pc1_0 packed B-operand stream (tile-pair layout)
//   PACK_P2: 65536 floats pc2 packed
//   PACK_P1: 16384 floats pc1_1 packed
// Packed layout: tile pair p (columns 32p..32p+31), lane l, 4 floats:
//   half = l>>4, lan = l&15
//   half0: {x,y} of point 16*(2p)+lan   then {x,y} of point 16*(2p+1)+lan
//   half1: {z,n2} of those same two points
// ---------------------------------------------------------------------------

static const size_t ACC_FLOATS = 16;
static const size_t PACK_A0_OFF = 16;                    // 65536 floats
static const size_t PACK_P2_OFF = 16 + 65536;            // 65536 floats
static const size_t PACK_P1_OFF = 16 + 2 * 65536;        // 16384 floats
static const size_t WS_FLOATS_NEEDED = 16 + 2 * 65536 + 16384;

__device__ __forceinline__ float group16_min(float m) {
    m = fminf(m, __shfl_xor(m, 1, 32));
    m = fminf(m, __shfl_xor(m, 2, 32));
    m = fminf(m, __shfl_xor(m, 4, 32));
    m = fminf(m, __shfl_xor(m, 8, 32));
    return m;
}

__device__ __forceinline__ float wave32_sum(float s) {
    s += __shfl_xor(s, 1, 32);
    s += __shfl_xor(s, 2, 32);
    s += __shfl_xor(s, 4, 32);
    s += __shfl_xor(s, 8, 32);
    s += __shfl_xor(s, 16, 32);
    return s;
}

// --- column-operand packing ------------------------------------------------
__global__ void pack_cols_kernel(const float* __restrict__ P, int n,
                                 float* __restrict__ out) {
    const int i = blockIdx.x * blockDim.x + threadIdx.x;
    if (i >= n) return;
    const float x = P[3 * i], y = P[3 * i + 1], z = P[3 * i + 2];
    const float n2 = x * x + y * y + z * z;
    const int t = i >> 4, lan = i & 15;
    const int p = t >> 1, sub = t & 1;
    float* base = out + (size_t)p * 128;  // 32 lanes * 4 floats
    base[lan * 4 + sub * 2 + 0] = x;
    base[lan * 4 + sub * 2 + 1] = y;
    base[(lan + 16) * 4 + sub * 2 + 0] = z;
    base[(lan + 16) * 4 + sub * 2 + 1] = n2;
}

// --- A-side operand construction (hoisted, per wave) -----------------------
__device__ __forceinline__ void make_a_operand(const float* __restrict__ A,
                                               int r0, int lan, int half,
                                               v2f& aop, float& mynorm2) {
    const float* ap = A + (size_t)(r0 + lan) * 3;
    const float ax = ap[0], ay = ap[1], az = ap[2];
    aop.x = half ? (-2.0f * az) : (-2.0f * ax);
    aop.y = half ? 1.0f : (-2.0f * ay);
    // ||a||^2 for the row THIS lane finalizes: row r0 + lan + 8*half
    const float* np = A + (size_t)(r0 + lan + 8 * half) * 3;
    const float nx = np[0], ny = np[1], nz = np[2];
    mynorm2 = nx * nx + ny * ny + nz * nz;
}

// --- chamfer min-rows, packed columns --------------------------------------
__global__ void chamfer_minrows_packed(const float* __restrict__ A,
                                       const float* __restrict__ packedB,
                                       int Nb, float* __restrict__ accum) {
    const int lane = threadIdx.x & 31;
    const int wid = threadIdx.x >> 5;
    const int rowTile = blockIdx.x * (blockDim.x >> 5) + wid;
    const int r0 = rowTile * 16;
    const int half = lane >> 4;
    const int lan = lane & 15;

    v2f aop; float mynorm2;
    make_a_operand(A, r0, lan, half, aop, mynorm2);

    float rowmin[8];
#pragma unroll
    for (int v = 0; v < 8; ++v) rowmin[v] = 3.0e38f;

    const v4f* bptr = (const v4f*)packedB + lane;
    for (int c0 = 0; c0 < Nb; c0 += 32) {
        const v4f b2 = bptr[(size_t)(c0 >> 5) * 32];
        const v2f b0 = {b2.x, b2.y};
        const v2f b1 = {b2.z, b2.w};
        const v8f c = {0.f, 0.f, 0.f, 0.f, 0.f, 0.f, 0.f, 0.f};
        const v8f d0 = __builtin_amdgcn_wmma_f32_16x16x4_f32(
            false, aop, false, b0, (short)0, c, false, false);
        const v8f d1 = __builtin_amdgcn_wmma_f32_16x16x4_f32(
            false, aop, false, b1, (short)0, c, false, false);
#pragma unroll
        for (int v = 0; v < 8; ++v)
            rowmin[v] = fminf(rowmin[v], fminf(d0[v], d1[v]));
    }

    float local = 0.0f;
#pragma unroll
    for (int v = 0; v < 8; ++v) {
        const float m = group16_min(rowmin[v]);
        if (lan == v) local += sqrtf(fmaxf(m + mynorm2, 0.0f));
    }
    local = wave32_sum(local);
    if (lane == 0) atomicAdd(accum, local);
}

// --- batched confidence term, packed pc2 columns ---------------------------
__global__ void confidence_packed(const float* __restrict__ pc3,
                                  const float* __restrict__ packedP2,
                                  const float* __restrict__ pc1_3,
                                  float* __restrict__ accum) {
    const int lane = threadIdx.x & 31;
    const int wid = threadIdx.x >> 5;
    const int rowTile = blockIdx.x * (blockDim.x >> 5) + wid;  // 0..127
    const int b = rowTile >> 4;
    const int r0 = (rowTile & 15) * 16;
    const int half = lane >> 4;
    const int lan = lane & 15;

    const float* A = pc3 + (size_t)b * 256 * 3;

    v2f aop; float mynorm2;
    make_a_operand(A, r0, lan, half, aop, mynorm2);

    float rowmin[8];
#pragma unroll
    for (int v = 0; v < 8; ++v) rowmin[v] = 3.0e38f;

    // batch b columns: global tile-pairs b*64 .. b*64+63
    const v4f* bptr = (const v4f*)packedP2 + (size_t)b * 64 * 32 + lane;
    for (int c0 = 0; c0 < 2048; c0 += 32) {
        const v4f b2 = bptr[(size_t)(c0 >> 5) * 32];
        const v2f b0 = {b2.x, b2.y};
        const v2f b1 = {b2.z, b2.w};
        const v8f c = {0.f, 0.f, 0.f, 0.f, 0.f, 0.f, 0.f, 0.f};
        const v8f d0 = __builtin_amdgcn_wmma_f32_16x16x4_f32(
            false, aop, false, b0, (short)0, c, false, false);
        const v8f d1 = __builtin_amdgcn_wmma_f32_16x16x4_f32(
            false, aop, false, b1, (short)0, c, false, false);
#pragma unroll
        for (int v = 0; v < 8; ++v)
            rowmin[v] = fminf(rowmin[v], fminf(d0[v], d1[v]));
    }

    float local = 0.0f;
#pragma unroll
    for (int v = 0; v < 8; ++v) {
        const float m = group16_min(rowmin[v]);
        if (lan == v) {
            const float gt = expf(-sqrtf(fmaxf(m + mynorm2, 0.0f)));
            const int rowLocal = r0 + v + 8 * half;
            const float diff = pc1_3[(size_t)b * 256 + rowLocal] - gt;
            local += diff * diff;
        }
    }
    local = wave32_sum(local);
    if (lane == 0) atomicAdd(accum, local);
}

// --- fallback (self-contained, no workspace packing) -----------------------
__global__ void chamfer_minrows_kernel(const float* __restrict__ A,
                                       const float* __restrict__ B,
                                       int Nb, float* __restrict__ accum) {
    const int lane = threadIdx.x & 31;
    const int wid = threadIdx.x >> 5;
    const int rowTile = blockIdx.x * (blockDim.x >> 5) + wid;
    const int r0 = rowTile * 16;
    const int half = lane >> 4;
    const int lan = lane & 15;

    v2f aop; float mynorm2;
    make_a_operand(A, r0, lan, half, aop, mynorm2);

    float rowmin[8];
#pragma unroll
    for (int v = 0; v < 8; ++v) rowmin[v] = 3.0e38f;

    for (int c0 = 0; c0 < Nb; c0 += 16) {
        const float* bp = B + (size_t)(c0 + lan) * 3;
        const float bx = bp[0], by = bp[1], bz = bp[2];
        v2f bop;
        bop.x = half ? bz : bx;
        bop.y = half ? (bx * bx + by * by + bz * bz) : by;
        const v8f c = {0.f, 0.f, 0.f, 0.f, 0.f, 0.f, 0.f, 0.f};
        const v8f d = __builtin_amdgcn_wmma_f32_16x16x4_f32(
            false, aop, false, bop, (short)0, c, false, false);
#pragma unroll
        for (int v = 0; v < 8; ++v) rowmin[v] = fminf(rowmin[v], d[v]);
    }

    float local = 0.0f;
#pragma unroll
    for (int v = 0; v < 8; ++v) {
        const float m = group16_min(rowmin[v]);
        if (lan == v) local += sqrtf(fmaxf(m + mynorm2, 0.0f));
    }
    local = wave32_sum(local);
    if (lane == 0) atomicAdd(accum, local);
}

__global__ void confidence_kernel(const float* __restrict__ pc3,
                                  const float* __restrict__ pc2,
                                  const float* __restrict__ pc1_3,
                                  float* __restrict__ accum) {
    const int lane = threadIdx.x & 31;
    const int wid = threadIdx.x >> 5;
    const int rowTile = blockIdx.x * (blockDim.x >> 5) + wid;
    const int b = rowTile >> 4;
    const int r0 = (rowTile & 15) * 16;
    const int half = lane >> 4;
    const int lan = lane & 15;

    const float* A = pc3 + (size_t)b * 256 * 3;
    const float* B = pc2 + (size_t)b * 2048 * 3;

    v2f aop; float mynorm2;
    make_a_operand(A, r0, lan, half, aop, mynorm2);

    float rowmin[8];
#pragma unroll
    for (int v = 0; v < 8; ++v) rowmin[v] = 3.0e38f;

    for (int c0 = 0; c0 < 2048; c0 += 16) {
        const float* bp = B + (size_t)(c0 + lan) * 3;
        const float bx = bp[0], by = bp[1], bz = bp[2];
        v2f bop;
        bop.x = half ? bz : bx;
        bop.y = half ? (bx * bx + by * by + bz * bz) : by;
        const v8f c = {0.f, 0.f, 0.f, 0.f, 0.f, 0.f, 0.f, 0.f};
        const v8f d = __builtin_amdgcn_wmma_f32_16x16x4_f32(
            false, aop, false, bop, (short)0, c, false, false);
#pragma unroll
        for (int v = 0; v < 8; ++v) rowmin[v] = fminf(rowmin[v], d[v]);
    }

    float local = 0.0f;
#pragma unroll
    for (int v = 0; v < 8; ++v) {
        const float m = group16_min(rowmin[v]);
        if (lan == v) {
            const float gt = expf(-sqrtf(fmaxf(m + mynorm2, 0.0f)));
            const float diff = pc1_3[(size_t)b * 256 + r0 + v + 8 * half] - gt;
            local += diff * diff;
        }
    }
    local = wave32_sum(local);
    if (lane == 0) atomicAdd(accum, local);
}

// --- misc ------------------------------------------------------------------
__global__ void p2p_kernel(const float* __restrict__ x,
                           const float* __restrict__ y,
                           int n, float* __restrict__ accum) {
    float s = 0.0f;
    for (int i = blockIdx.x * blockDim.x + threadIdx.x; i < n;
         i += gridDim.x * blockDim.x) {
        const float d = x[i] - y[i];
        s += d * d;
    }
    s = wave32_sum(s);
    if ((threadIdx.x & 31) == 0) atomicAdd(accum, s);
}

__global__ void zero_accum_kernel(float* __restrict__ accum) {
    if (threadIdx.x < 8) accum[threadIdx.x] = 0.0f;
}

__global__ void finalize_kernel(const float* __restrict__ accum,
                                float* __restrict__ out) {
    const float cd = (accum[0] + accum[1]) * (1.0f / 16384.0f);
    const float seed = accum[2] * (1.0f / 4096.0f) + accum[3] * (1.0f / 16384.0f);
    const float conf = accum[4] * (1.0f / 2048.0f);
    const float p2p = accum[5] * (1.0f / 49152.0f);
    out[0] = 0.5f * cd + 0.5f * seed + 0.5f * conf + p2p;
}

extern "C" void kernel_launch(void* const* d_in, const int* in_sizes, int n_in,
                              void* d_out, int out_size, void* d_ws, size_t ws_size,
                              hipStream_t stream) {
    const float* pc1_0 = (const float*)d_in[0];  // [8,2048,3]
    const float* pc1_1 = (const float*)d_in[1];  // [8, 512,3]
    const float* pc1_3 = (const float*)d_in[2];  // [8, 256,1]
    const float* pc2   = (const float*)d_in[3];  // [8,2048,3]
    const float* pc3   = (const float*)d_in[4];  // [8, 256,3]
    float* out = (float*)d_out;
    float* ws = (float*)d_ws;
    float* accum = ws;

    (void)in_sizes; (void)n_in; (void)out_size;

    zero_accum_kernel<<<1, 32, 0, stream>>>(accum);

    const bool usePacked = ws_size >= WS_FLOATS_NEEDED * sizeof(float);
    if (usePacked) {
        float* packA0 = ws + PACK_A0_OFF;
        float* packP2 = ws + PACK_P2_OFF;
        float* packP1 = ws + PACK_P1_OFF;
        pack_cols_kernel<<<64, 256, 0, stream>>>(pc1_0, 16384, packA0);
        pack_cols_kernel<<<64, 256, 0, stream>>>(pc2, 16384, packP2);
        pack_cols_kernel<<<16, 256, 0, stream>>>(pc1_1, 4096, packP1);

        chamfer_minrows_packed<<<256, 128, 0, stream>>>(pc1_0, packP2, 16384, accum + 0);
        chamfer_minrows_packed<<<256, 128, 0, stream>>>(pc2, packA0, 16384, accum + 1);
        chamfer_minrows_packed<<<64, 128, 0, stream>>>(pc1_1, packP2, 16384, accum + 2);
        chamfer_minrows_packed<<<256, 128, 0, stream>>>(pc2, packP1, 4096, accum + 3);
        confidence_packed<<<32, 128, 0, stream>>>(pc3, packP2, pc1_3, accum + 4);
    } else {
        chamfer_minrows_kernel<<<256, 128, 0, stream>>>(pc1_0, pc2, 16384, accum + 0);
        chamfer_minrows_kernel<<<256, 128, 0, stream>>>(pc2, pc1_0, 16384, accum + 1);
        chamfer_minrows_kernel<<<64, 128, 0, stream>>>(pc1_1, pc2, 16384, accum + 2);
        chamfer_minrows_kernel<<<256, 128, 0, stream>>>(pc2, pc1_1, 4096, accum + 3);
        confidence_kernel<<<32, 128, 0, stream>>>(pc3, pc2, pc1_3, accum + 4);
    }

    p2p_kernel<<<96, 256, 0, stream>>>(pc1_0, pc2, 49152, accum + 5);
    finalize_kernel<<<1, 1, 0, stream>>>(accum, out);
}